// OptimizedGNNPredictor_67886253081017
// MI455X (gfx1250) — compile-verified
//
#include <hip/hip_runtime.h>

// CDNA5 / gfx1250. wave32. fp32 WMMA (16x16x4) for the dense GEMMs; the
// graph aggregation (SpMM) is the memory-bound part and runs out of L2
// (hs + acc buffers are 51 MB total, well under the 192 MB L2).

typedef __attribute__((ext_vector_type(2))) float v2f;
typedef __attribute__((ext_vector_type(8))) float v8f;

// ---------------- degree / normalization ----------------

__global__ void deg_init_kernel(float* __restrict__ deg, int n) {
  int i = blockIdx.x * blockDim.x + threadIdx.x;
  if (i < n) deg[i] = 1.0f;  // self-loop contributes 1 to every node's degree
}

__global__ void deg_accum_kernel(const int* __restrict__ dst, float* __restrict__ deg, int e) {
  int i = blockIdx.x * blockDim.x + threadIdx.x;
  if (i < e) atomicAdd(&deg[dst[i]], 1.0f);
}

__global__ void deg_to_dinv_kernel(float* __restrict__ deg, int n) {
  int i = blockIdx.x * blockDim.x + threadIdx.x;
  if (i < n) {
    float d = deg[i];
    deg[i] = (d > 0.0f) ? rsqrtf(d) : 0.0f;  // deg >= 1 always (self-loops)
  }
}

// ---------------- fused GEMM + row-scale epilogue ----------------
// Computes hs = (X @ W) * dinv[row]  and  acc = hs  (self-loop init fused).
// X: [nrows, 128] row-major, W: [128, NOUT] row-major.
// One wave per 16x16 output tile; (NOUT/16) waves per block, one M-tile/block.

template <int NOUT>
__global__ __launch_bounds__((NOUT / 16) * 32)
void gemm_norm_kernel(const float* __restrict__ X, const float* __restrict__ W,
                      const float* __restrict__ dinv,
                      float* __restrict__ hs, float* __restrict__ acc, int nrows) {
  constexpr int K = 128;
  constexpr int NTHREADS = (NOUT / 16) * 32;
  __shared__ float sA[16 * K];  // 8 KB row tile of X, shared by all N-tiles

  const int tid = threadIdx.x;
  const int mbase = blockIdx.x * 16;

  for (int i = tid; i < 16 * K; i += NTHREADS) {
    int r = i >> 7;           // /128
    int c = i & (K - 1);
    int gr = mbase + r;
    sA[i] = (gr < nrows) ? X[(size_t)gr * K + c] : 0.0f;
  }
  __syncthreads();

  const int wave = tid >> 5;
  const int lane = tid & 31;
  const int half = lane >> 4;   // 0: lanes 0-15, 1: lanes 16-31
  const int l15  = lane & 15;
  const int nbase = wave * 16;

  v8f c = {};
  #pragma unroll
  for (int kk = 0; kk < K; kk += 4) {
    // A 16x4 fp32: VGPR0 = {K=kk (lanes0-15), K=kk+2 (lanes16-31)}, VGPR1 = +1
    v2f a, b;
    a.x = sA[l15 * K + kk + 2 * half];
    a.y = sA[l15 * K + kk + 2 * half + 1];
    // B 4x16 fp32: VGPR0 = row {kk | kk+2}, VGPR1 = row {kk+1 | kk+3}
    b.x = W[(size_t)(kk + 2 * half) * NOUT + nbase + l15];
    b.y = W[(size_t)(kk + 2 * half + 1) * NOUT + nbase + l15];
    c = __builtin_amdgcn_wmma_f32_16x16x4_f32(false, a, false, b, (short)0, c,
                                              false, false);
  }

  // C/D layout: VGPR r -> M = r (lanes 0-15) / r+8 (lanes 16-31), N = lane&15
  const int moff = half * 8;
  #pragma unroll
  for (int r = 0; r < 8; ++r) {
    int gm = mbase + r + moff;
    if (gm < nrows) {
      float v = c[r] * dinv[gm];
      size_t idx = (size_t)gm * NOUT + nbase + l15;
      hs[idx]  = v;   // gather source for SpMM
      acc[idx] = v;   // self-loop edge (i,i) contributes hs[i] to acc[i]
    }
  }
}

// ---------------- edge scatter (SpMM): acc[dst] += hs[src] ----------------
// One wave32 per edge; lane moves F/32 consecutive floats.
// All regions are 16B-aligned by construction (see workspace layout), so use
// explicit vector loads to get global_load_b128/b64.

template <int F>
__global__ void scatter_kernel(const int* __restrict__ src, const int* __restrict__ dst,
                               const float* __restrict__ hs, float* __restrict__ acc,
                               int ne) {
  constexpr int PER = F / 32;  // 4 (F=128) or 2 (F=64)
  const int wid = (blockIdx.x * blockDim.x + threadIdx.x) >> 5;
  const int lane = threadIdx.x & 31;
  if (wid >= ne) return;
  const int s = src[wid];   // wave-uniform
  const int d = dst[wid];
  const float* hp = hs + (size_t)s * F + lane * PER;
  float* ap = acc + (size_t)d * F + lane * PER;
  if constexpr (PER == 4) {
    float4 v = *(const float4*)hp;          // global_load_b128
    atomicAdd(&ap[0], v.x);
    atomicAdd(&ap[1], v.y);
    atomicAdd(&ap[2], v.z);
    atomicAdd(&ap[3], v.w);
  } else {
    float2 v = *(const float2*)hp;          // global_load_b64
    atomicAdd(&ap[0], v.x);
    atomicAdd(&ap[1], v.y);
  }
}

// ---------------- out = relu(acc * dinv[dst] + bias), float4-vectorized ----

template <int F>
__global__ void bias_relu_kernel(const float* __restrict__ acc, const float* __restrict__ dinv,
                                 const float* __restrict__ bias, float* __restrict__ out,
                                 int n) {
  int i = blockIdx.x * blockDim.x + threadIdx.x;  // one float4 per thread
  if (i >= n * (F / 4)) return;
  int base = i * 4;
  int node = base >> (F == 128 ? 7 : 6);
  int f = base & (F - 1);
  float4 a = *(const float4*)(acc + base);
  float4 b = *(const float4*)(bias + f);
  float di = dinv[node];
  float4 v;
  v.x = fmaxf(a.x * di + b.x, 0.0f);
  v.y = fmaxf(a.y * di + b.y, 0.0f);
  v.z = fmaxf(a.z * di + b.z, 0.0f);
  v.w = fmaxf(a.w * di + b.w, 0.0f);
  *(float4*)(out + base) = v;
}

// ---------------- driver ----------------

extern "C" void kernel_launch(void* const* d_in, const int* in_sizes, int n_in,
                              void* d_out, int out_size, void* d_ws, size_t ws_size,
                              hipStream_t stream) {
  const float* x  = (const float*)d_in[0];
  const int*   ei = (const int*)d_in[1];
  const float* W1 = (const float*)d_in[2];
  const float* b1 = (const float*)d_in[3];
  const float* W2 = (const float*)d_in[4];
  const float* b2 = (const float*)d_in[5];

  const int N = in_sizes[0] / 128;  // 50000
  const int E = in_sizes[1] / 2;    // 800000
  const int* src = ei;
  const int* dst = ei + E;

  // Workspace layout (floats): dinv[N] | hs1[N*128] | acc1[N*128]
  // N*4 bytes = 200000 ≡ 0 (mod 16), N*128*4 and N*64*4 are 16B multiples,
  // so every region below is 16B-aligned (d_ws itself is allocation-aligned).
  // Layer-2 reuses: h1 -> hs1 region; hs2/acc2 -> acc1 region (each N*64).
  float* ws   = (float*)d_ws;
  float* dinv = ws;
  float* hs1  = dinv + N;
  float* acc1 = hs1 + (size_t)N * 128;
  float* h1   = hs1;
  float* hs2  = acc1;
  float* acc2 = acc1 + (size_t)N * 64;
  float* out  = (float*)d_out;

  // Degrees -> dinv (in place)
  deg_init_kernel<<<(N + 255) / 256, 256, 0, stream>>>(dinv, N);
  deg_accum_kernel<<<(E + 255) / 256, 256, 0, stream>>>(dst, dinv, E);
  deg_to_dinv_kernel<<<(N + 255) / 256, 256, 0, stream>>>(dinv, N);

  const int mtiles = (N + 15) / 16;  // 3125
  const int eblocks = (E + 7) / 8;   // 8 waves (edges) per 256-thread block

  // ---- Layer 1: 128 -> 128 ----
  gemm_norm_kernel<128><<<mtiles, 256, 0, stream>>>(x, W1, dinv, hs1, acc1, N);
  scatter_kernel<128><<<eblocks, 256, 0, stream>>>(src, dst, hs1, acc1, E);
  bias_relu_kernel<128><<<(N * 32 + 255) / 256, 256, 0, stream>>>(acc1, dinv, b1, h1, N);

  // ---- Layer 2: 128 -> 64 ----
  gemm_norm_kernel<64><<<mtiles, 128, 0, stream>>>(h1, W2, dinv, hs2, acc2, N);
  scatter_kernel<64><<<eblocks, 256, 0, stream>>>(src, dst, hs2, acc2, E);
  bias_relu_kernel<64><<<(N * 16 + 255) / 256, 256, 0, stream>>>(acc2, dinv, b2, out, N);
}